// NONLocalBlock2D_40192303956656
// MI455X (gfx1250) — compile-verified
//
#include <hip/hip_runtime.h>
#include <math.h>
#include <stdint.h>

// ---- problem constants (fixed by reference setup_inputs) ----
#define Bk    64
#define Ck    256
#define C2k   128
#define Tk    64
#define Vk    25
#define Nk    1600     // T*V
#define Sk    725      // (1+4+8+16)*V
#define SPADk 768      // Sk padded to 48*16: full WMMA tiles AND 48%8==0 so
                       // 8-wave blocks cover s-tiles with EXEC all-ones

typedef __attribute__((ext_vector_type(2))) float v2f;
typedef __attribute__((ext_vector_type(8))) float v8f;

__device__ __forceinline__ v8f wmma4(v2f a, v2f b, v8f c) {
  // D = A(16x4) * B(4x16) + C, exact fp32
  return __builtin_amdgcn_wmma_f32_16x16x4_f32(false, a, false, b, (short)0, c,
                                               false, false);
}

// Async copy 16B/lane global -> LDS (ASYNCcnt-tracked, no VGPR data movement).
// ISA 15.18.3 opcode 98: VDST = per-lane LDS byte address, VADDR = 64b global.
__device__ __forceinline__ void async_g2l_b128(const float* gsrc, const float* lds) {
  const uint32_t laddr = (uint32_t)(uintptr_t)lds;  // flat->LDS: low 32 bits
  asm volatile("global_load_async_to_lds_b128 %0, %1, off"
               :: "v"(laddr), "v"(gsrc) : "memory");
}
__device__ __forceinline__ void async_wait0() {
  asm volatile("s_wait_asynccnt 0x0" ::: "memory");
}

// ---------------------------------------------------------------------------
// Kernel 1: g = g_w @ x + g_b ; theta = theta_w @ x + theta_b
// 16 waves/block share one async-staged x tile (256K x 16N = 16 KB in LDS);
// wave w: branch = w>>3 (g/theta), oc tile = (w&7)*16. x read ONCE from HBM.
// ---------------------------------------------------------------------------
__global__ void __launch_bounds__(512)
k_conv_gtheta(const float* __restrict__ x,
              const float* __restrict__ g_w, const float* __restrict__ g_b,
              const float* __restrict__ th_w, const float* __restrict__ th_b,
              float* __restrict__ g_out, float* __restrict__ th_out) {
  __shared__ float xt[Ck * 16];                 // [k][n], 16-float rows
  const int tid = threadIdx.x;
  const int n0 = blockIdx.x * 16;
  const int b  = blockIdx.z;
  const float* X = x + (size_t)b * Ck * Nk;

#pragma unroll
  for (int i = 0; i < 2; ++i) {                 // 1024 b128 lane-ops / 512 thr
    const int idx = tid + i * 512;
    const int row = idx >> 2, q = idx & 3;      // row = k, q*4 = n offset
    async_g2l_b128(X + (size_t)row * Nk + n0 + q * 4, &xt[row * 16 + q * 4]);
  }
  async_wait0();
  __syncthreads();

  const int wave = tid >> 5;
  const int lane = tid & 31, half = lane >> 4, l16 = lane & 15;
  const int branch = wave >> 3;
  const int m0 = (wave & 7) * 16;
  const float* W  = branch ? th_w : g_w;        // (C2, C)
  const float* Bs = branch ? th_b : g_b;

  v8f acc = {};
#pragma unroll 4
  for (int k0 = 0; k0 < Ck; k0 += 4) {
    const int ka = k0 + 2 * half;
    v2f a, bb;
    a.x = W[(m0 + l16) * Ck + ka];
    a.y = W[(m0 + l16) * Ck + ka + 1];
    bb.x = xt[ka * 16 + l16];                   // ds_load, conflict-free
    bb.y = xt[(ka + 1) * 16 + l16];
    acc = wmma4(a, bb, acc);
  }

  if (branch == 0) {
    float* G = g_out + (size_t)b * C2k * Nk;    // (B, C2, N) for PSP reshape
#pragma unroll
    for (int r = 0; r < 8; ++r) {
      const int m = m0 + r + 8 * half;
      G[(size_t)m * Nk + n0 + l16] = acc[r] + Bs[m];
    }
  } else {
    float* Th = th_out + (size_t)b * Nk * C2k;  // (B, N, C2) transposed
#pragma unroll
    for (int r = 0; r < 8; ++r) {
      const int m = m0 + r + 8 * half;
      Th[(size_t)(n0 + l16) * C2k + m] = acc[r] + Bs[m];
    }
  }
}

// ---------------------------------------------------------------------------
// Kernel 2: PSP multi-scale pool, faithful to torch's RAW view/reshape chain.
// Writes phi (B,C2,SPAD) and gp^T (B,SPAD,C2), zero-filling cols [Sk,SPADk).
// ---------------------------------------------------------------------------
__global__ void k_psp(const float* __restrict__ g_flat,
                      float* __restrict__ phi, float* __restrict__ gp) {
  const int tid = blockIdx.x * blockDim.x + threadIdx.x;
  const int total = Bk * C2k * SPADk;
  if (tid >= total) return;
  const int kt = tid % SPADk;
  const int cp = (tid / SPADk) % C2k;
  const int b  = tid / (SPADk * C2k);

  float val = 0.0f;
  if (kt < Sk) {
    int s, k;
    if (kt < 25)       { s = 1;  k = kt;       }
    else if (kt < 125) { s = 4;  k = kt - 25;  }
    else if (kt < 325) { s = 8;  k = kt - 125; }
    else               { s = 16; k = kt - 325; }
    const int flat = b * (C2k * s * Vk) + cp * (s * Vk) + k;   // p raw index
    const int m   = flat / (C2k * s);
    const int rem = flat % (C2k * s);
    const int ch  = rem / s;
    const int si  = rem % s;
    const int L = Tk / s;
    const float* src = g_flat + (size_t)(m * C2k + ch) * Tk + si * L;
    float sum = 0.0f;
    for (int t = 0; t < L; ++t) sum += src[t];
    val = sum / (float)L;
  }
  phi[(size_t)b * C2k * SPADk + (size_t)cp * SPADk + kt] = val;
  gp [(size_t)b * SPADk * C2k + (size_t)kt * C2k + cp]  = val;
}

// ---------------------------------------------------------------------------
// Kernel 3: f[b,n,s] = sum_c theta[b,n,c] * phi[b,c,s]  (M=1600,N=768,K=128)
// 8 waves/block share one async-staged theta tile (16x128, stride-132 padded
// => fragment reads hit 32 distinct LDS banks); wave w -> s-tile y*8+w.
// ---------------------------------------------------------------------------
__global__ void __launch_bounds__(256)
k_fgemm(const float* __restrict__ th, const float* __restrict__ phi,
        float* __restrict__ f) {
  __shared__ float at[16 * 132];                // [n][k] padded
  const int tid = threadIdx.x;
  const int n0 = blockIdx.x * 16;
  const int b  = blockIdx.z;
  const float* A = th + (size_t)b * Nk * C2k;

#pragma unroll
  for (int i = 0; i < 2; ++i) {                 // 512 b128 lane-ops / 256 thr
    const int idx = tid + i * 256;
    const int row = idx >> 5, q = idx & 31;
    async_g2l_b128(A + (size_t)(n0 + row) * C2k + q * 4,
                   &at[row * 132 + q * 4]);
  }
  async_wait0();
  __syncthreads();

  const int wave = tid >> 5;
  const int lane = tid & 31, half = lane >> 4, l16 = lane & 15;
  const int s0 = (blockIdx.y * 8 + wave) * 16;
  const float* P = phi + (size_t)b * C2k * SPADk;

  v8f acc = {};
#pragma unroll 4
  for (int k0 = 0; k0 < C2k; k0 += 4) {
    const int ka = k0 + 2 * half;
    v2f a, bb;
    a.x = at[l16 * 132 + ka];
    a.y = at[l16 * 132 + ka + 1];
    bb.x = P[(size_t)ka * SPADk + s0 + l16];
    bb.y = P[(size_t)(ka + 1) * SPADk + s0 + l16];
    acc = wmma4(a, bb, acc);
  }
  float* F = f + (size_t)b * Nk * SPADk;
#pragma unroll
  for (int r = 0; r < 8; ++r)
    F[(size_t)(n0 + r + 8 * half) * SPADk + s0 + l16] = acc[r];
}

// ---------------------------------------------------------------------------
// Kernel 4/6: online column softmax stats (max, sum-exp) over rows, axis -2.
// ---------------------------------------------------------------------------
__global__ void k_colstats(const float* __restrict__ src, int ld, int nrows,
                           float* __restrict__ omax, float* __restrict__ osum) {
  const int c = blockIdx.x * blockDim.x + threadIdx.x;
  const int b = blockIdx.y;
  if (c >= ld) return;
  const float* S = src + (size_t)b * nrows * ld;
  float m = -3.402823466e+38f, z = 0.0f;
  for (int n = 0; n < nrows; ++n) {
    const float v = S[(size_t)n * ld + c];
    if (v > m) { z = z * __expf(m - v) + 1.0f; m = v; }
    else       { z += __expf(v - m); }
  }
  omax[(size_t)b * ld + c] = m;
  osum[(size_t)b * ld + c] = z;
}

// ---------------------------------------------------------------------------
// Kernel 5: y[b,n,c] = sum_s softmax_n(f)[n,s] * gp[s,c]
// f (HBM-resident, 314 MB) is async-staged ONCE per n-tile (16x768 = 48 KB,
// stride-772 padded) and shared by 8 waves covering all 8 c-tiles; softmax
// (exp(.-m_s), 1/Z_s) fused into the fragment builders.
// ---------------------------------------------------------------------------
__global__ void __launch_bounds__(256)
k_ygemm(const float* __restrict__ f, const float* __restrict__ gp,
        const float* __restrict__ cm, const float* __restrict__ cs,
        float* __restrict__ y) {
  __shared__ float ft[16 * 772];                // 49408 B
  const int tid = threadIdx.x;
  const int n0 = blockIdx.x * 16;
  const int b  = blockIdx.z;
  const float* F = f + (size_t)b * Nk * SPADk;

#pragma unroll
  for (int i = 0; i < 12; ++i) {                // 3072 b128 lane-ops / 256 thr
    const int idx = tid + i * 256;
    const int row = idx / 192, q = idx % 192;   // 192 quads per 768-col row
    async_g2l_b128(F + (size_t)(n0 + row) * SPADk + q * 4,
                   &ft[row * 772 + q * 4]);
  }
  async_wait0();
  __syncthreads();

  const int wave = tid >> 5;
  const int lane = tid & 31, half = lane >> 4, l16 = lane & 15;
  const int c0 = wave * 16;
  const float* GP = gp + (size_t)b * SPADk * C2k;
  const float* CM = cm + (size_t)b * SPADk;
  const float* CS = cs + (size_t)b * SPADk;

  v8f acc = {};
  for (int k0 = 0; k0 < SPADk; k0 += 4) {
    const int ka = k0 + 2 * half;
    v2f a, bb;
    a.x = __expf(ft[l16 * 772 + ka]     - CM[ka]);
    a.y = __expf(ft[l16 * 772 + ka + 1] - CM[ka + 1]);
    bb.x = GP[(size_t)ka * C2k + c0 + l16]       / CS[ka];
    bb.y = GP[(size_t)(ka + 1) * C2k + c0 + l16] / CS[ka + 1];
    acc = wmma4(a, bb, acc);
  }
  float* Y = y + (size_t)b * Nk * C2k;
#pragma unroll
  for (int r = 0; r < 8; ++r)
    Y[(size_t)(n0 + r + 8 * half) * C2k + c0 + l16] = acc[r];
}

// ---------------------------------------------------------------------------
// Kernel 7: out = relu(BN(W_w @ softmax_n(y) + W_b + x))
// 16 waves/block (one per 16-oc tile) share an async-staged raw y tile
// (16x128, stride-132); softmax-2 fused into B-fragment build; full epilogue.
// ---------------------------------------------------------------------------
__global__ void __launch_bounds__(512)
k_outgemm(const float* __restrict__ y, const float* __restrict__ cmy,
          const float* __restrict__ csy, const float* __restrict__ Ww,
          const float* __restrict__ Wb, const float* __restrict__ x,
          const float* __restrict__ gamma, const float* __restrict__ beta,
          const float* __restrict__ mean, const float* __restrict__ var,
          float* __restrict__ out) {
  __shared__ float yt[16 * 132];
  const int tid = threadIdx.x;
  const int n0 = blockIdx.x * 16;
  const int b  = blockIdx.z;
  const float* Y = y + (size_t)b * Nk * C2k;

  {                                             // 512 b128 lane-ops / 512 thr
    const int row = tid >> 5, q = tid & 31;
    async_g2l_b128(Y + (size_t)(n0 + row) * C2k + q * 4,
                   &yt[row * 132 + q * 4]);
  }
  async_wait0();
  __syncthreads();

  const int wave = tid >> 5;
  const int lane = tid & 31, half = lane >> 4, l16 = lane & 15;
  const int o0 = wave * 16;
  const float* CMY = cmy + (size_t)b * C2k;
  const float* CSY = csy + (size_t)b * C2k;

  v8f acc = {};
#pragma unroll 2
  for (int k0 = 0; k0 < C2k; k0 += 4) {
    const int ka = k0 + 2 * half;
    v2f a, bb;
    a.x = Ww[(o0 + l16) * C2k + ka];
    a.y = Ww[(o0 + l16) * C2k + ka + 1];
    bb.x = __expf(yt[l16 * 132 + ka]     - CMY[ka])     / CSY[ka];
    bb.y = __expf(yt[l16 * 132 + ka + 1] - CMY[ka + 1]) / CSY[ka + 1];
    acc = wmma4(a, bb, acc);
  }

  const float* X = x + (size_t)b * Ck * Nk;
  float* O = out + (size_t)b * Ck * Nk;
#pragma unroll
  for (int r = 0; r < 8; ++r) {
    const int o = o0 + r + 8 * half;
    const int n = n0 + l16;
    const float z = acc[r] + Wb[o] + X[(size_t)o * Nk + n];
    const float inv = rsqrtf(var[o] + 1e-5f);
    const float v = (z - mean[o]) * (inv * gamma[o]) + beta[o];
    O[(size_t)o * Nk + n] = v > 0.0f ? v : 0.0f;
  }
}

// ---------------------------------------------------------------------------
extern "C" void kernel_launch(void* const* d_in, const int* in_sizes, int n_in,
                              void* d_out, int out_size, void* d_ws,
                              size_t ws_size, hipStream_t stream) {
  const float* x    = (const float*)d_in[0];
  const float* g_w  = (const float*)d_in[1];
  const float* g_b  = (const float*)d_in[2];
  const float* th_w = (const float*)d_in[3];
  const float* th_b = (const float*)d_in[4];
  const float* W_w  = (const float*)d_in[5];
  const float* W_b  = (const float*)d_in[6];
  const float* bn_g = (const float*)d_in[7];
  const float* bn_b = (const float*)d_in[8];
  const float* bn_m = (const float*)d_in[9];
  const float* bn_v = (const float*)d_in[10];
  float* out = (float*)d_out;

  float* ws     = (float*)d_ws;
  float* g_buf  = ws;                                   // B*C2*N
  float* th_buf = g_buf + (size_t)Bk * C2k * Nk;        // B*N*C2
  float* phi    = th_buf + (size_t)Bk * Nk * C2k;       // B*C2*SPAD
  float* gp     = phi + (size_t)Bk * C2k * SPADk;       // B*SPAD*C2
  float* f_buf  = gp + (size_t)Bk * SPADk * C2k;        // B*N*SPAD (~315 MB)
  float* cmf    = f_buf + (size_t)Bk * Nk * SPADk;      // B*SPAD
  float* csf    = cmf + (size_t)Bk * SPADk;             // B*SPAD
  float* y_buf  = csf + (size_t)Bk * SPADk;             // B*N*C2
  float* cmy    = y_buf + (size_t)Bk * Nk * C2k;        // B*C2
  float* csy    = cmy + (size_t)Bk * C2k;               // B*C2

  // 1) g & theta convs: x tile async-staged once, 16 WMMA waves share it
  k_conv_gtheta<<<dim3(Nk / 16, 1, Bk), 512, 0, stream>>>(
      x, g_w, g_b, th_w, th_b, g_buf, th_buf);

  // 2) PSP pooling -> phi / gp^T (zero-padded to SPAD)
  {
    const int total = Bk * C2k * SPADk;
    k_psp<<<(total + 255) / 256, 256, 0, stream>>>(g_buf, phi, gp);
  }

  // 3) f = theta @ phi (theta tile async-staged, 8 s-tiles per block)
  k_fgemm<<<dim3(Nk / 16, SPADk / 16 / 8, Bk), 256, 0, stream>>>(
      th_buf, phi, f_buf);

  // 4) column softmax stats over f (axis -2)
  k_colstats<<<dim3(SPADk / 256, Bk), 256, 0, stream>>>(
      f_buf, SPADk, Nk, cmf, csf);

  // 5) y = softmax_n(f) @ gp : f read from HBM exactly once (async -> LDS)
  k_ygemm<<<dim3(Nk / 16, 1, Bk), 256, 0, stream>>>(
      f_buf, gp, cmf, csf, y_buf);

  // 6) column softmax stats over y
  k_colstats<<<dim3(1, Bk), C2k, 0, stream>>>(y_buf, C2k, Nk, cmy, csy);

  // 7) final conv + residual + BN + ReLU (softmax-2 fused, y staged once)
  k_outgemm<<<dim3(Nk / 16, 1, Bk), 512, 0, stream>>>(
      y_buf, cmy, csy, W_w, W_b, x, bn_g, bn_b, bn_m, bn_v, out);

  (void)in_sizes; (void)n_in; (void)out_size; (void)ws_size;
}